// UniformBottomUpHTMM_87900800680702
// MI455X (gfx1250) — compile-verified
//
#include <hip/hip_runtime.h>

typedef __attribute__((ext_vector_type(2))) float v2f;
typedef __attribute__((ext_vector_type(8))) float v8f;

#define TDEPTH  9
#define NPT     1023      // nodes per tree
#define NGEN    8
#define NTREES  64
#define BLK     256
#define NWAVE   (BLK / 32)

enum : int {
  OFF_SMAP  = 0,                    // smA zero-padded [16][16]      256
  OFF_ALOGA = OFF_SMAP  + 256,      // smA*logA zero-padded [8][16]  128
  OFF_PI    = OFF_ALOGA + 128,      // sm_Pi[i]                        8
  OFF_LPI   = OFF_PI    + 8,        // log sm_Pi[i]                    8
  OFF_KC    = OFF_LPI   + 8,        // rowmax+log(rowsum)              8
  OFF_PART  = OFF_KC    + 8,        // reduction scratch             256
  OFF_LB    = OFF_PART  + 256,      // log B_x [1023][8]
  OFF_BETA  = OFF_LB    + NPT * 8,  // beta    [1023][8]
  OFF_TBETA = OFF_BETA  + NPT * 8,  // t_beta  [511][8] (internal only)
  OFF_EPS   = OFF_TBETA + 511 * 8,  // eps     [1023][8]
  OFF_SS    = OFF_EPS   + NPT * 8,  // S  scratch [256][8]
  OFF_SSL   = OFF_SS    + 256 * 8,  // SL scratch [256][8]
  OFF_RED   = OFF_SSL   + 256 * 8,  // block reduce [256]
  OFF_X     = OFF_RED   + 256,      // x gather [1023] (ints)
  SMEM_FLOATS = OFF_X + NPT         // 34679 floats = 138716 B (<160KB -> 2 WG/WGP)
};

__device__ __forceinline__ v8f wmma4(v2f a, v2f b, v8f c) {
  // D(16x16,f32) = A(16x4,f32) x B(4x16,f32) + C   — exact f32 math
  return __builtin_amdgcn_wmma_f32_16x16x4_f32(false, a, false, b, (short)0, c,
                                               false, false);
}

__global__ __launch_bounds__(BLK)
void htmm_fused_kernel(const float* __restrict__ Ag, const float* __restrict__ Bg,
                       const float* __restrict__ Pig, const int* __restrict__ xg,
                       float* __restrict__ out) {
  extern __shared__ float sm[];
  float* smAp  = sm + OFF_SMAP;   // [16][16], rows/cols >=8 are zero
  float* AlogA = sm + OFF_ALOGA;  // [8][16],  cols >=8 are zero
  float* pis   = sm + OFF_PI;
  float* lpis  = sm + OFF_LPI;
  float* Kc    = sm + OFF_KC;
  float* part  = sm + OFF_PART;
  float* lb    = sm + OFF_LB;     // lb[n*8+i] = log B_x ; B_x = expf(lb)
  float* bet   = sm + OFF_BETA;
  float* tbeta = sm + OFF_TBETA;
  float* epsv  = sm + OFF_EPS;
  float* sS    = sm + OFF_SS;
  float* sSL   = sm + OFF_SSL;
  float* red   = sm + OFF_RED;
  int*   xloc  = (int*)(sm + OFF_X);

  const int tid  = threadIdx.x;
  const int lane = tid & 31;
  const int wid  = tid >> 5;          // 8 waves per block
  const int g    = blockIdx.x & 7;    // generative head
  const int t    = blockIdx.x >> 3;   // tree

  // ---------------- phase A: zero pads, x gather, B row-max partials ----------
  for (int idx = tid; idx < 256 + 128; idx += BLK) smAp[idx] = 0.f; // covers AlogA too
  for (int n = tid; n < NPT; n += BLK) xloc[n] = xg[t * NPT + n];
  {
    const int r = tid >> 5, sl = tid & 31;
    float mx = -1e30f;
    for (int m = sl; m < 256; m += 32) mx = fmaxf(mx, Bg[(r * 256 + m) * 8 + g]);
    part[tid] = mx;
  }
  __syncthreads();

  // ---------------- phase B: A/Pi softmaxes + row-max reduce ------------------
  if (tid < 8) {
    { // row max reduce
      float mx = -1e30f;
      for (int k = 0; k < 32; ++k) mx = fmaxf(mx, part[tid * 32 + k]);
      Kc[tid] = mx;                   // temporarily holds rowmax
    }
    { // column j = tid of A: softmax over parent dim i (axis 0)
      const int j = tid;
      float v[8], mx = -1e30f;
      for (int i = 0; i < 8; ++i) { v[i] = Ag[(i * 8 + j) * 8 + g]; mx = fmaxf(mx, v[i]); }
      float s = 0.f;
      for (int i = 0; i < 8; ++i) s += expf(v[i] - mx);
      const float ls = logf(s), inv = 1.f / s;
      for (int i = 0; i < 8; ++i) {
        float a = expf(v[i] - mx) * inv;
        smAp[i * 16 + j]  = a;
        AlogA[i * 16 + j] = a * (v[i] - mx - ls);
      }
    }
  } else if (tid == 8) {              // Pi softmax over i
    float v[8], mx = -1e30f;
    for (int i = 0; i < 8; ++i) { v[i] = Pig[i * 8 + g]; mx = fmaxf(mx, v[i]); }
    float s = 0.f;
    for (int i = 0; i < 8; ++i) s += expf(v[i] - mx);
    const float ls = logf(s), inv = 1.f / s;
    for (int i = 0; i < 8; ++i) { pis[i] = expf(v[i] - mx) * inv; lpis[i] = v[i] - mx - ls; }
  }
  __syncthreads();

  // ---------------- phase C/D: B row exp-sums -> Kc = rowmax + log(rowsum) ----
  {
    const int r = tid >> 5, sl = tid & 31;
    const float rm = Kc[r];
    float s = 0.f;
    for (int m = sl; m < 256; m += 32) s += expf(Bg[(r * 256 + m) * 8 + g] - rm);
    part[tid] = s;
  }
  __syncthreads();
  if (tid < 8) {
    float s = 0.f;
    for (int k = 0; k < 32; ++k) s += part[tid * 32 + k];
    Kc[tid] = Kc[tid] + logf(s);
  }
  __syncthreads();

  // ---------------- phase E: gathered log B_x -------------------------------
  for (int idx = tid; idx < NPT * 8; idx += BLK) {
    const int n = idx >> 3, i = idx & 7;
    lb[idx] = Bg[(i * 256 + xloc[n]) * 8 + g] - Kc[i];
  }
  __syncthreads();

  // ---------------- phase 1: leaves beta ------------------------------------
  for (int n = 511 + tid; n < NPT; n += BLK) {
    float tv[8], s = 0.f;
    for (int i = 0; i < 8; ++i) { float v = pis[i] * expf(lb[n * 8 + i]); tv[i] = v; s += v; }
    const float inv = 1.f / s;
    for (int i = 0; i < 8; ++i) bet[n * 8 + i] = tv[i] * inv;
  }
  __syncthreads();

  const int mrow = lane & 15;         // A row M / B col N / D col N
  const int hi   = lane >> 4;         // 0/1 -> K split per ISA A/B layout

  // B operands are lane-only functions: hoist out of ALL level loops.
  // Upward:  Bmat[j][i] = smA[i][j]  -> smAp[mrow*16 + j], zero for mrow>=8
  // Down S:  Bmat[i][j] = smA[i][j]  -> smAp[i*16 + mrow], zero for mrow>=8
  // Down SL: Bmat[i][j] = (A*logA)   -> AlogA[i*16 + mrow]
  v2f bUp[2], bS[2], bSL[2];
#pragma unroll
  for (int chunk = 0; chunk < 2; ++chunk) {
    const int k0 = chunk * 4 + hi * 2;
    bUp[chunk].x = smAp[mrow * 16 + k0 + 0];
    bUp[chunk].y = smAp[mrow * 16 + k0 + 1];
    bS [chunk].x = smAp[(k0 + 0) * 16 + mrow];
    bS [chunk].y = smAp[(k0 + 1) * 16 + mrow];
    bSL[chunk].x = AlogA[(k0 + 0) * 16 + mrow];
    bSL[chunk].y = AlogA[(k0 + 1) * 16 + mrow];
  }

  // ---------------- phase 2: upward sweep (WMMA f32 16x16x4) ----------------
  for (int l = TDEPTH; l >= 1; --l) {
    const int E = 1 << l;             // children at this level
    const int cbase = E - 1, pbase = (E >> 1) - 1;
    const int ntile = (E + 15) >> 4;
    for (int tile = wid; tile < ntile; tile += NWAVE) {  // wave-uniform
      const int coff0 = tile * 16;
      const int node  = cbase + coff0 + mrow;            // always in-bounds;
      v8f acc = {0.f, 0.f, 0.f, 0.f, 0.f, 0.f, 0.f, 0.f};
#pragma unroll
      for (int chunk = 0; chunk < 2; ++chunk) {          // K = 8 = 2 x 4
        const int k0 = chunk * 4 + hi * 2;
        v2f a;                        // garbage rows only feed unstored D rows
        a.x = bet[node * 8 + k0 + 0];
        a.y = bet[node * 8 + k0 + 1];
        acc = wmma4(a, bUp[chunk], acc);  // D[m][i] = sum_j smA[i][j]*beta[m][j]
      }
      // D row pairs (2q,2q+1) -> parent mean (t_beta = sum/2)
      if (E - coff0 >= 16) {          // full tile: one exec region
        if (mrow < 8) {
#pragma unroll
          for (int q = 0; q < 4; ++q) {
            const int poff = (coff0 >> 1) + q + hi * 4;
            tbeta[(pbase + poff) * 8 + mrow] = 0.5f * (acc[2 * q] + acc[2 * q + 1]);
          }
        }
      } else {                        // partial tile (levels with E<16)
        const int Ecnt = E - coff0;
        if (mrow < 8) {
#pragma unroll
          for (int q = 0; q < 4; ++q) {
            const int m = 2 * q + hi * 8;
            if (m < Ecnt) {
              const int poff = (coff0 >> 1) + q + hi * 4;
              tbeta[(pbase + poff) * 8 + mrow] = 0.5f * (acc[2 * q] + acc[2 * q + 1]);
            }
          }
        }
      }
    }
    __syncthreads();
    const int P = E >> 1;             // parents: beta = normalize(t_beta * B_x)
    for (int p = tid; p < P; p += BLK) {
      const int node = pbase + p;
      float tv[8], s = 0.f;
      for (int i = 0; i < 8; ++i) {
        float v = tbeta[node * 8 + i] * expf(lb[node * 8 + i]); tv[i] = v; s += v;
      }
      const float inv = 1.f / s;
      for (int i = 0; i < 8; ++i) bet[node * 8 + i] = tv[i] * inv;
    }
    __syncthreads();
  }

  // ---------------- phase 3: downward sweep + likelihood --------------------
  float acc_ll = 0.f;
  if (tid == 0) {                     // root: eps = beta ; node term eps.log(B_x)
    float s = 0.f;
    for (int i = 0; i < 8; ++i) { float e = bet[i]; epsv[i] = e; s += e * lb[i]; }
    acc_ll += s;
  }
  __syncthreads();
  for (int l = 1; l <= TDEPTH; ++l) {
    const int P = 1 << (l - 1);       // parents at level l-1
    const int pbase = P - 1, cbase = 2 * P - 1;
    const int ntile = (P + 15) >> 4;
    for (int tile = wid; tile < ntile; tile += NWAVE) {  // wave-uniform
      const int poff0 = tile * 16;
      const int node  = pbase + poff0 + mrow;            // in-bounds for all levels
      v8f aS  = {0.f, 0.f, 0.f, 0.f, 0.f, 0.f, 0.f, 0.f};
      v8f aSL = {0.f, 0.f, 0.f, 0.f, 0.f, 0.f, 0.f, 0.f};
#pragma unroll
      for (int chunk = 0; chunk < 2; ++chunk) {
        const int k0 = chunk * 4 + hi * 2;               // contraction index i
        v2f a;                        // w[p][i]; garbage rows feed unstored D rows
        a.x = epsv[node * 8 + k0 + 0] / tbeta[node * 8 + k0 + 0];
        a.y = epsv[node * 8 + k0 + 1] / tbeta[node * 8 + k0 + 1];
        aS  = wmma4(a, bS[chunk],  aS);   // S[p][j]  = sum_i w*smA
        aSL = wmma4(a, bSL[chunk], aSL);  // SL[p][j] = sum_i w*smA*logA
      }
      if (P - poff0 >= 16) {          // full tile
        if (mrow < 8) {
#pragma unroll
          for (int r = 0; r < 8; ++r) {
            const int m = r + hi * 8;
            sS [(poff0 + m) * 8 + mrow] = aS[r];
            sSL[(poff0 + m) * 8 + mrow] = aSL[r];
          }
        }
      } else {                        // partial tile
        const int Pcnt = P - poff0;
        if (mrow < 8) {
#pragma unroll
          for (int r = 0; r < 8; ++r) {
            const int m = r + hi * 8;
            if (m < Pcnt) {
              sS [(poff0 + m) * 8 + mrow] = aS[r];
              sSL[(poff0 + m) * 8 + mrow] = aSL[r];
            }
          }
        }
      }
    }
    __syncthreads();
    const int E = 2 * P;
    const bool isleaf = (l == TDEPTH);
    for (int c = tid; c < E; c += BLK) {
      const int cn = cbase + c, po = c >> 1;
      float s = 0.f;
      for (int j = 0; j < 8; ++j) {
        const float bj = bet[cn * 8 + j];
        const float e  = bj * sS[po * 8 + j];            // eps[ch][j]
        epsv[cn * 8 + j] = e;
        s += e * lb[cn * 8 + j]                          // eps . log B_x  (all nodes)
           + 0.5f * bj * sSL[po * 8 + j];                // t_eps . log A  (per edge /2)
        if (isleaf) s += e * lpis[j];                    // eps . log Pi   (leaves)
      }
      acc_ll += s;
    }
    __syncthreads();
  }

  // ---------------- phase 4: block reduction, write -ll ---------------------
  red[tid] = acc_ll;
  __syncthreads();
  for (int off = BLK / 2; off > 0; off >>= 1) {
    if (tid < off) red[tid] += red[tid + off];
    __syncthreads();
  }
  if (tid == 0) out[t * 8 + g] = -red[0];
}

extern "C" void kernel_launch(void* const* d_in, const int* in_sizes, int n_in,
                              void* d_out, int out_size, void* d_ws, size_t ws_size,
                              hipStream_t stream) {
  (void)in_sizes; (void)n_in; (void)out_size; (void)d_ws; (void)ws_size;
  const float* A  = (const float*)d_in[0];
  const float* B  = (const float*)d_in[1];
  const float* Pi = (const float*)d_in[2];
  const int*   x  = (const int*)d_in[3];
  float* out = (float*)d_out;
  const size_t shmem = (size_t)SMEM_FLOATS * sizeof(float);   // ~139 KB
  htmm_fused_kernel<<<dim3(NTREES * NGEN), dim3(BLK), shmem, stream>>>(A, B, Pi, x, out);
}